// My_CrossEntropyLoss_2705829396890
// MI455X (gfx1250) — compile-verified
//
#include <hip/hip_runtime.h>
#include <hip/hip_bf16.h>

// Cross-entropy pick-log-mean:  out = -(1/B) * sum_i log(pred[i, label[i]])
// Pass 1: per-block async label stage (global->LDS, ASYNCcnt) + scattered
//         NT gather + wave32 tree-reduce -> partial sums in d_ws
// Pass 2: single block reduces the partials -> d_out[0]
// Fully deterministic (no float atomics; fixed reduction tree).

#define CE_BLOCK 256

__device__ __forceinline__ float block_reduce_sum(float v, float* s_red) {
    // wave32 tree reduction
    #pragma unroll
    for (int off = 16; off > 0; off >>= 1)
        v += __shfl_down(v, off, 32);
    const int lane = threadIdx.x & 31;
    const int wave = threadIdx.x >> 5;
    if (lane == 0) s_red[wave] = v;
    __syncthreads();
    const int nwaves = (blockDim.x + 31) >> 5;
    v = (threadIdx.x < (unsigned)nwaves) ? s_red[threadIdx.x] : 0.0f;
    if (wave == 0) {
        #pragma unroll
        for (int off = 16; off > 0; off >>= 1)
            v += __shfl_down(v, off, 32);
    }
    return v;  // valid in thread 0
}

__global__ void ce_gather_partial(const float* __restrict__ pred,
                                  const int* __restrict__ label,
                                  float* __restrict__ partial,
                                  int B, long long C) {
    __shared__ int   s_label[CE_BLOCK];
    __shared__ float s_red[CE_BLOCK / 32];

    const int tid = blockIdx.x * blockDim.x + threadIdx.x;

    float t = 0.0f;
    if (tid < B) {
        // --- CDNA5 async global->LDS stage of this block's label slice ---
        // Per-lane: copies label[tid] from global memory to s_label[threadIdx.x] in LDS (ASYNCcnt-tracked)
        unsigned long long gaddr =
            (unsigned long long)(size_t)(label + tid);
        unsigned lds_addr =
            (unsigned)(size_t)(&s_label[threadIdx.x]);  // low 32 bits = LDS offset
        asm volatile("global_load_async_to_lds_b32 %0, %1, off"
                     :
                     : "v"(lds_addr), "v"(gaddr)
                     : "memory");
        asm volatile("s_wait_asynccnt 0" ::: "memory");

        const int l = s_label[threadIdx.x];  // each lane reads its own slot

        // The single scattered gather that dominates this kernel.
        // Single-touch data: non-temporal so we don't pollute WGP$/L2.
        const float v =
            __builtin_nontemporal_load(&pred[(size_t)tid * (size_t)C + (size_t)l]);

        // v in (1e-6, 1]: always normal, so use the bare v_log_f32 path
        // (skips the denormal pre-scale cmp/cndmask/ldexp of __logf).
        t = __builtin_amdgcn_logf(v) * 0.6931471805599453f;  // log2(v) * ln(2)
    }
    __syncthreads();  // all waves reach the reduction together

    const float s = block_reduce_sum(t, s_red);
    if (threadIdx.x == 0) partial[blockIdx.x] = s;
}

__global__ void ce_finalize(const float* __restrict__ partial, int nparts,
                            float* __restrict__ out, float invB) {
    __shared__ float s_red[CE_BLOCK / 32];
    float v = 0.0f;
    for (int i = threadIdx.x; i < nparts; i += blockDim.x)
        v += partial[i];
    const float s = block_reduce_sum(v, s_red);
    if (threadIdx.x == 0) out[0] = -s * invB;
}

extern "C" void kernel_launch(void* const* d_in, const int* in_sizes, int n_in,
                              void* d_out, int out_size, void* d_ws, size_t ws_size,
                              hipStream_t stream) {
    const float* pred  = (const float*)d_in[0];
    const int*   label = (const int*)d_in[1];

    const int       B = in_sizes[1];
    const long long C = (long long)in_sizes[0] / (long long)B;

    float* partial = (float*)d_ws;

    const int threads = CE_BLOCK;
    const int blocks  = (B + threads - 1) / threads;

    ce_gather_partial<<<blocks, threads, 0, stream>>>(pred, label, partial, B, C);
    ce_finalize<<<1, CE_BLOCK, 0, stream>>>(partial, blocks, (float*)d_out,
                                            1.0f / (float)B);
}